// PredecessorDecoder_59304908423638
// MI455X (gfx1250) — compile-verified
//
#include <hip/hip_runtime.h>
#include <hip/hip_bf16.h>

// Problem constants (match reference)
#define NN 8192      // nodes
#define EE 524288    // edges
#define DD 256       // latent dim
#define NEG_FILL -1e9f

typedef __attribute__((ext_vector_type(2))) float v2f;
typedef __attribute__((ext_vector_type(4))) float v4f;
typedef __attribute__((ext_vector_type(8))) float v8f;

// ---------------------------------------------------------------------------
// Kernel 1: fill the 8192x8192 output with -1e9 using non-temporal B128 stores
// (268 MB of pure store traffic; NT hint keeps A/B resident in L2)
// ---------------------------------------------------------------------------
__global__ __launch_bounds__(256) void fill_kernel(v4f* __restrict__ out, int n4) {
    v4f v = {NEG_FILL, NEG_FILL, NEG_FILL, NEG_FILL};
    int stride = gridDim.x * blockDim.x;
    for (int i = blockIdx.x * blockDim.x + threadIdx.x; i < n4; i += stride) {
        __builtin_nontemporal_store(v, &out[i]);
    }
}

// ---------------------------------------------------------------------------
// Kernel 2: A = h @ W1[0:256,:], B = h @ W1[256:512,:]  (fp32 WMMA 16x16x4)
// One wave per 16x16 output tile; K-loop of 64 WMMA ops.
//   A-frag (16x4 f32): lane l holds rows M=l%16, K = 2*(l/16)+{0,1} in v[0..1]
//   B-frag (4x16 f32): lane l holds col  N=l%16, K = 2*(l/16)+{0,1} in v[0..1]
//   C/D (16x16 f32):   VGPR r, lane l -> M = r + 8*(l/16), N = l%16
// ---------------------------------------------------------------------------
__global__ __launch_bounds__(32) void gemm_hW_kernel(const float* __restrict__ h,
                                                     const float* __restrict__ W1,
                                                     float* __restrict__ AB) {
    const int TM = NN / 16;   // 512 row tiles
    const int TN = DD / 16;   // 16 col tiles
    int bx  = blockIdx.x;               // [0, 2*TM*TN)
    int mat = bx / (TM * TN);           // 0 -> A (W1 rows 0..255), 1 -> B (rows 256..511)
    int rem = bx % (TM * TN);
    int tm  = rem / TN;
    int tn  = rem % TN;

    int lane = threadIdx.x & 31;
    int lm   = lane & 15;               // M (A) / N (B) index within tile
    int kh   = lane >> 4;               // K half selector

    int m0 = tm * 16;
    int n0 = tn * 16;
    const float* __restrict__ Wb = W1 + (size_t)mat * 256 * DD;  // row block of W1

    v8f c = {};
    #pragma unroll 4
    for (int k0 = 0; k0 < DD; k0 += 4) {
        int k = k0 + 2 * kh;
        v2f a = *(const v2f*)(h + (size_t)(m0 + lm) * DD + k);   // two consecutive K
        v2f b;
        b.x = Wb[(size_t)(k + 0) * DD + n0 + lm];
        b.y = Wb[(size_t)(k + 1) * DD + n0 + lm];
        // 8 args: (neg_a, A, neg_b, B, c_mod, C, reuse_a, reuse_b)
        c = __builtin_amdgcn_wmma_f32_16x16x4_f32(false, a, false, b,
                                                  (short)0, c, false, false);
    }

    float* __restrict__ outp = AB + (size_t)mat * NN * DD;
    int mrow = m0 + 8 * kh;
    int ncol = n0 + lm;
    #pragma unroll
    for (int r = 0; r < 8; ++r) {
        outp[(size_t)(mrow + r) * DD + ncol] = c[r];
    }
}

// ---------------------------------------------------------------------------
// Kernel 3: per-edge score + scatter. One wave32 per edge (grid-stride).
// Lane t owns channels j = 8t..8t+7; wave-reduce via shfl_xor; lane 0 stores.
// All A/B row gathers hit L2 (A+B = 16 MB << 192 MB L2).
// ---------------------------------------------------------------------------
__global__ __launch_bounds__(256) void edge_kernel(const float* __restrict__ A,
                                                   const float* __restrict__ B,
                                                   const int*   __restrict__ ei,
                                                   const float* __restrict__ ea,
                                                   const float* __restrict__ W1,
                                                   const float* __restrict__ W2,
                                                   float* __restrict__ out) {
    int tid    = blockIdx.x * blockDim.x + threadIdx.x;
    int wave   = tid >> 5;
    int lane   = tid & 31;
    int nwaves = (gridDim.x * blockDim.x) >> 5;
    int jb     = lane * 8;

    // Invariant per-channel weights: last W1 row + W2 column (reused every edge)
    v4f wl0 = *(const v4f*)(W1 + (size_t)512 * DD + jb);
    v4f wl1 = *(const v4f*)(W1 + (size_t)512 * DD + jb + 4);
    v4f w20 = *(const v4f*)(W2 + jb);
    v4f w21 = *(const v4f*)(W2 + jb + 4);

    for (int e = wave; e < EE; e += nwaves) {
        int   l = ei[e];
        int   r = ei[EE + e];
        float t = ea[e];

        const v4f* ap = (const v4f*)(A + (size_t)l * DD + jb);
        const v4f* bp = (const v4f*)(B + (size_t)r * DD + jb);
        v4f a0 = ap[0], a1 = ap[1];
        v4f b0 = bp[0], b1 = bp[1];

        float acc;
        acc  = fmaxf(fmaf(t, wl0.x, a0.x + b0.x), 0.f) * w20.x;
        acc += fmaxf(fmaf(t, wl0.y, a0.y + b0.y), 0.f) * w20.y;
        acc += fmaxf(fmaf(t, wl0.z, a0.z + b0.z), 0.f) * w20.z;
        acc += fmaxf(fmaf(t, wl0.w, a0.w + b0.w), 0.f) * w20.w;
        acc += fmaxf(fmaf(t, wl1.x, a1.x + b1.x), 0.f) * w21.x;
        acc += fmaxf(fmaf(t, wl1.y, a1.y + b1.y), 0.f) * w21.y;
        acc += fmaxf(fmaf(t, wl1.z, a1.z + b1.z), 0.f) * w21.z;
        acc += fmaxf(fmaf(t, wl1.w, a1.w + b1.w), 0.f) * w21.w;

        // wave32 all-reduce
        #pragma unroll
        for (int o = 16; o > 0; o >>= 1)
            acc += __shfl_xor(acc, o, 32);

        if (lane == 0)
            out[(size_t)l * NN + r] = acc;
    }
}

// ---------------------------------------------------------------------------
// kernel_launch: fill -> GEMM (WMMA) -> edge scatter, all stream-ordered.
// d_in order: encoded (unused), h, edge_index, edge_attr, W1, W2
// d_ws layout: A [NN*DD floats] then B [NN*DD floats] = 16 MB
// ---------------------------------------------------------------------------
extern "C" void kernel_launch(void* const* d_in, const int* in_sizes, int n_in,
                              void* d_out, int out_size, void* d_ws, size_t ws_size,
                              hipStream_t stream) {
    const float* h  = (const float*)d_in[1];
    const int*   ei = (const int*)d_in[2];
    const float* ea = (const float*)d_in[3];
    const float* W1 = (const float*)d_in[4];
    const float* W2 = (const float*)d_in[5];
    float* out = (float*)d_out;

    float* A = (float*)d_ws;            // NN*DD floats
    float* B = A + (size_t)NN * DD;     // NN*DD floats

    // 1) Fill output with -1e9 (16 M float4 stores)
    int n4 = (NN * (int)NN) / 4;        // 16,777,216
    fill_kernel<<<2048, 256, 0, stream>>>((v4f*)out, n4);

    // 2) A = h@W1_top, B = h@W1_mid : 2 * 512 * 16 tiles, one wave each
    gemm_hW_kernel<<<2 * (NN / 16) * (DD / 16), 32, 0, stream>>>(h, W1, A);

    // 3) Per-edge MLP tail + scatter (16384 waves, 32 edges each)
    edge_kernel<<<2048, 256, 0, stream>>>(A, B, ei, ea, W1, W2, out);
}